// TriplaneEncoding_18605798326298
// MI455X (gfx1250) — compile-verified
//
#include <hip/hip_runtime.h>
#include <stdint.h>

#define RES 512
#define RR  (RES * RES)
#define NCH 64
#define NPL 3

// ---------------------------------------------------------------------------
// Pass 1: transpose plane_coef (3, 64, 512, 512) -> planesT (3, 512*512, 64)
// LDS-tiled: 64 pixels x 64 channels per block, padded to avoid bank conflicts.
// ---------------------------------------------------------------------------
__global__ __launch_bounds__(256) void tp_transpose(const float* __restrict__ src,
                                                    float* __restrict__ dst) {
    __shared__ float tile[64][65];
    const int plane   = blockIdx.x >> 12;         // / 4096
    const int pixBase = (blockIdx.x & 4095) << 6; // tile * 64
    const int t = threadIdx.x;
    const int q = t >> 6;   // 0..3
    const int r = t & 63;   // 0..63

#pragma unroll
    for (int i = 0; i < 16; ++i) {
        const int c = i * 4 + q; // channel 0..63, bijective
        // coalesced read: consecutive lanes -> consecutive pixels
        tile[r][c] = src[(size_t)(plane * NCH + c) * RR + pixBase + r];
    }
    __syncthreads();
#pragma unroll
    for (int i = 0; i < 16; ++i) {
        const int pl = i * 4 + q; // local pixel 0..63
        // coalesced write: consecutive lanes -> consecutive channels
        dst[((size_t)plane * RR + pixBase + pl) * NCH + r] = tile[pl][r];
    }
}

// ---------------------------------------------------------------------------
// Per-point sampling math from pre-loaded coords: 12 corner-row addresses
// (already offset by this lane's float4 slot) + 12 bilinear weights.
// Row j = plane*4 + corner; corner order matches the reference:
//   0:(y0,x0) (1-wx)(1-wy)  1:(y0,x1) wx(1-wy)  2:(y1,x0) (1-wx)wy  3:(y1,x1) wx*wy
// ---------------------------------------------------------------------------
__device__ __forceinline__ void tp_point_setup(float ax, float ay, float az,
                                               int sub, const float* __restrict__ planesT,
                                               uint64_t va[12], float w[12]) {
    const float U[3] = {ax, ax, ay};
    const float V[3] = {ay, az, az};
#pragma unroll
    for (int pl = 0; pl < NPL; ++pl) {
        float fx = fminf(fmaxf((U[pl] + 1.0f) * 0.5f * (float)(RES - 1), 0.0f), (float)(RES - 1));
        float fy = fminf(fmaxf((V[pl] + 1.0f) * 0.5f * (float)(RES - 1), 0.0f), (float)(RES - 1));
        const int x0 = (int)fx;                 // fx >= 0, trunc == floor
        const int y0 = (int)fy;
        const float wx = fx - (float)x0;
        const float wy = fy - (float)y0;
        const int x1 = (x0 + 1 < RES) ? x0 + 1 : RES - 1;
        const int y1 = (y0 + 1 < RES) ? y0 + 1 : RES - 1;
        const size_t pb = (size_t)pl * RR;
        const float* r00 = planesT + (pb + (size_t)y0 * RES + x0) * NCH + sub * 4;
        const float* r01 = planesT + (pb + (size_t)y0 * RES + x1) * NCH + sub * 4;
        const float* r10 = planesT + (pb + (size_t)y1 * RES + x0) * NCH + sub * 4;
        const float* r11 = planesT + (pb + (size_t)y1 * RES + x1) * NCH + sub * 4;
        va[pl * 4 + 0] = (uint64_t)(uintptr_t)r00;
        va[pl * 4 + 1] = (uint64_t)(uintptr_t)r01;
        va[pl * 4 + 2] = (uint64_t)(uintptr_t)r10;
        va[pl * 4 + 3] = (uint64_t)(uintptr_t)r11;
        w[pl * 4 + 0] = (1.0f - wx) * (1.0f - wy);
        w[pl * 4 + 1] = wx * (1.0f - wy);
        w[pl * 4 + 2] = (1.0f - wx) * wy;
        w[pl * 4 + 3] = wx * wy;
    }
}

// ---------------------------------------------------------------------------
// Pass 2: gather + bilinear combine, double-buffered through LDS with the
// CDNA5 async-copy engine (global_load_async_to_lds_b128 / s_wait_asynccnt),
// with the point-coordinate loads software-pipelined one extra stage ahead.
// 16-lane half-wave owns one point; lane owns 4 channels (float4).
// ---------------------------------------------------------------------------
#define BLOCK2 128
#define WAVES2 (BLOCK2 / 32)

__global__ __launch_bounds__(BLOCK2) void tp_gather_async(const float* __restrict__ pts,
                                                          const float* __restrict__ planesT,
                                                          float* __restrict__ out, int npts) {
    // 2 (double buffer) x 4 waves x 2 points x 12 rows x 64 floats = 48 KB
    __shared__ float buf[2][WAVES2][2][12][NCH];

    const int t    = threadIdx.x;
    const int wave = t >> 5;
    const int lane = t & 31;
    const int half = lane >> 4;  // which point within the wave
    const int sub  = lane & 15;  // float4 channel-group id

    const int hw0      = blockIdx.x * (BLOCK2 / 16) + (t >> 4);
    const int hwStride = gridDim.x * (BLOCK2 / 16);

    const uint32_t lds0 =
        (uint32_t)(uintptr_t)(&buf[0][wave][half][0][0]) + (uint32_t)sub * 16u;
    const uint32_t lds1 =
        (uint32_t)(uintptr_t)(&buf[1][wave][half][0][0]) + (uint32_t)sub * 16u;

    int p = hw0;
    if (p >= npts) return;

    // stage 0: load own coords (only unhidden point-load in the pipeline)
    uint64_t va[12];
    float    w[12];
    {
        const float ax = pts[3 * (size_t)p + 0];
        const float ay = pts[3 * (size_t)p + 1];
        const float az = pts[3 * (size_t)p + 2];
        tp_point_setup(ax, ay, az, sub, planesT, va, w);
    }
#pragma unroll
    for (int j = 0; j < 12; ++j) {
        const uint32_t la = lds0 + (uint32_t)j * 256u;
        asm volatile("global_load_async_to_lds_b128 %0, %1, off"
                     :: "v"(la), "v"(va[j]) : "memory");
    }

    // pre-load coords for stage 1 (consumed next iteration)
    float nx = 0.f, ny = 0.f, nz = 0.f;
    {
        const int p1 = p + hwStride;
        if (p1 < npts) {
            nx = pts[3 * (size_t)p1 + 0];
            ny = pts[3 * (size_t)p1 + 1];
            nz = pts[3 * (size_t)p1 + 2];
        }
    }

    int par = 0;
    for (;;) {
        const int  pn      = p + hwStride;
        const bool hasNext = (pn < npts);
        float w2[12];
        if (hasNext) {
            // addresses from coords pre-loaded a full stage ago (latency hidden)
            uint64_t va2[12];
            tp_point_setup(nx, ny, nz, sub, planesT, va2, w2);
            // make sure our previous LDS reads of the target buffer retired
            asm volatile("s_wait_dscnt 0" ::: "memory");
            const uint32_t base = (par == 0) ? lds1 : lds0;
#pragma unroll
            for (int j = 0; j < 12; ++j) {
                const uint32_t la = base + (uint32_t)j * 256u;
                asm volatile("global_load_async_to_lds_b128 %0, %1, off"
                             :: "v"(la), "v"(va2[j]) : "memory");
            }
            // pre-load coords for the stage after next; in flight during compute
            const int p2 = pn + hwStride;
            if (p2 < npts) {
                nx = pts[3 * (size_t)p2 + 0];
                ny = pts[3 * (size_t)p2 + 1];
                nz = pts[3 * (size_t)p2 + 2];
            }
            // 12 newly issued still in flight; current stage's 12 completed
            asm volatile("s_wait_asynccnt 12" ::: "memory");
        } else {
            asm volatile("s_wait_asynccnt 0" ::: "memory");
        }

        // combine current stage from LDS (ds_load_b128, conflict-free: the 16
        // lanes of a half-wave cover one contiguous 256B row)
        const float* bp = &buf[par][wave][half][0][sub * 4];
        float a0 = 0.f, a1 = 0.f, a2 = 0.f, a3 = 0.f;
#pragma unroll
        for (int j = 0; j < 12; ++j) {
            const float4 g = *(const float4*)(bp + (size_t)j * NCH);
            a0 = fmaf(w[j], g.x, a0);
            a1 = fmaf(w[j], g.y, a1);
            a2 = fmaf(w[j], g.z, a2);
            a3 = fmaf(w[j], g.w, a3);
        }
        float* op = out + (size_t)p * NCH + sub * 4;
        float4 acc;
        acc.x = a0; acc.y = a1; acc.z = a2; acc.w = a3;
        *(float4*)op = acc;   // global_store_b128, 256B per half-wave

        if (!hasNext) break;
        p   = pn;
        par ^= 1;
#pragma unroll
        for (int j = 0; j < 12; ++j) w[j] = w2[j];
    }
}

// ---------------------------------------------------------------------------
// Fallback (workspace too small for the transposed planes): direct gather from
// the original channel-major layout. Correct but lower line utilization.
// ---------------------------------------------------------------------------
__global__ __launch_bounds__(256) void tp_gather_direct(const float* __restrict__ pts,
                                                        const float* __restrict__ plane,
                                                        float* __restrict__ out, int npts) {
    const int hw  = (blockIdx.x * blockDim.x + threadIdx.x) >> 4;
    const int sub = threadIdx.x & 15;
    if (hw >= npts) return;

    const float ax = pts[3 * (size_t)hw + 0];
    const float ay = pts[3 * (size_t)hw + 1];
    const float az = pts[3 * (size_t)hw + 2];
    const float U[3] = {ax, ax, ay};
    const float V[3] = {ay, az, az};

    float a0 = 0.f, a1 = 0.f, a2 = 0.f, a3 = 0.f;
    const int c0 = sub * 4;
#pragma unroll
    for (int pl = 0; pl < NPL; ++pl) {
        float fx = fminf(fmaxf((U[pl] + 1.0f) * 0.5f * (float)(RES - 1), 0.0f), (float)(RES - 1));
        float fy = fminf(fmaxf((V[pl] + 1.0f) * 0.5f * (float)(RES - 1), 0.0f), (float)(RES - 1));
        const int x0 = (int)fx;
        const int y0 = (int)fy;
        const float wx = fx - (float)x0;
        const float wy = fy - (float)y0;
        const int x1 = (x0 + 1 < RES) ? x0 + 1 : RES - 1;
        const int y1 = (y0 + 1 < RES) ? y0 + 1 : RES - 1;
        const int   idx[4] = {y0 * RES + x0, y0 * RES + x1, y1 * RES + x0, y1 * RES + x1};
        const float ww[4]  = {(1.0f - wx) * (1.0f - wy), wx * (1.0f - wy),
                              (1.0f - wx) * wy,          wx * wy};
        const float* pb = plane + (size_t)pl * NCH * RR;
#pragma unroll
        for (int k = 0; k < 4; ++k) {
            const float* g = pb + idx[k];
            a0 = fmaf(ww[k], g[(size_t)(c0 + 0) * RR], a0);
            a1 = fmaf(ww[k], g[(size_t)(c0 + 1) * RR], a1);
            a2 = fmaf(ww[k], g[(size_t)(c0 + 2) * RR], a2);
            a3 = fmaf(ww[k], g[(size_t)(c0 + 3) * RR], a3);
        }
    }
    float* op = out + (size_t)hw * NCH + c0;
    op[0] = a0; op[1] = a1; op[2] = a2; op[3] = a3;
}

// ---------------------------------------------------------------------------
extern "C" void kernel_launch(void* const* d_in, const int* in_sizes, int n_in,
                              void* d_out, int out_size, void* d_ws, size_t ws_size,
                              hipStream_t stream) {
    const float* pts   = (const float*)d_in[0];   // (N, 3) f32
    const float* plane = (const float*)d_in[1];   // (3, 64, 512, 512) f32
    float*       out   = (float*)d_out;           // (N, 64) f32
    const int npts = in_sizes[0] / 3;

    const size_t need = (size_t)NPL * RR * NCH * sizeof(float); // 192 MiB
    if (ws_size >= need) {
        float* planesT = (float*)d_ws;
        tp_transpose<<<NPL * (RR / 64), 256, 0, stream>>>(plane, planesT);
        const int blocks2 = 8192; // 65536 half-waves -> 16 points each at N=1M
        tp_gather_async<<<blocks2, BLOCK2, 0, stream>>>(pts, planesT, out, npts);
    } else {
        const int threads = 256;
        const long long tot = (long long)npts * 16;
        const int blocks = (int)((tot + threads - 1) / threads);
        tp_gather_direct<<<blocks, threads, 0, stream>>>(pts, plane, out, npts);
    }
}